// CRF_16106127360636
// MI455X (gfx1250) — compile-verified
//
#include <hip/hip_runtime.h>
#include <stdint.h>

#define TAGS      64
#define SEQ_LEN   131072
#define CHUNK     128
#define NCHUNK    (SEQ_LEN / CHUNK)   /* 1024 */
#define START_IDX 1
#define STOP_IDX  2
#define NEGVAL    (-10000.0f)
#define NINF      (-1.0e30f)
#define FLOWEST   (-3.0e38f)

typedef float v2f __attribute__((ext_vector_type(2)));
typedef float v8f __attribute__((ext_vector_type(8)));
typedef unsigned int  u32x4 __attribute__((ext_vector_type(4)));
typedef int           i32x4 __attribute__((ext_vector_type(4)));
typedef int           i32x8 __attribute__((ext_vector_type(8)));
typedef unsigned char u8;

// ---------------------------------------------------------------------------
// K1: per-chunk max-plus aggregate matrix G_c[n][p] (includes in-chunk feats).
// G_new[n,p] = feat_t[n] + max_k( T[n,k] + G[k,p] );  G_0 = maxplus identity.
// 256 threads = 8 waves; wave pair covers 16 rows n, lanes carry columns p.
// G columns live in VGPRs; T rows streamed via wave-uniform (SGPR) loads.
// ---------------------------------------------------------------------------
__global__ __launch_bounds__(256) void k1_chunk_gemm(
    const float* __restrict__ feats, const float* __restrict__ trans,
    float* __restrict__ G) {
  const int c    = blockIdx.x;
  const int tid  = threadIdx.x;
  const int w    = __builtin_amdgcn_readfirstlane(tid >> 5);  // wave id: uniform
  const int lane = tid & 31;
  const int p    = ((w & 1) << 5) | lane;  // column 0..63
  const int n0   = (w >> 1) << 4;          // row-group base (uniform)

  __shared__ float Gbuf[2][TAGS][TAGS];    // double buffered, 32 KB

  float g[TAGS];                           // G[*][p] column in registers
#pragma unroll
  for (int k = 0; k < TAGS; ++k) g[k] = (k == p) ? 0.0f : NINF;

  const float* fbase = feats + (size_t)c * CHUNK * TAGS;

  for (int s = 0; s < CHUNK; ++s) {
    asm volatile("" ::: "memory");  // keep T row loads inside the step loop
    const int nb = s & 1;
    for (int i = 0; i < 16; ++i) {
      const int n = n0 + i;                         // wave-uniform row
      const float* __restrict__ trow = trans + n * TAGS;
      float m = NINF;
#pragma unroll
      for (int k = 0; k < TAGS; ++k) m = fmaxf(m, trow[k] + g[k]);
      Gbuf[nb][n][p] = m + fbase[s * TAGS + n];
    }
    __syncthreads();
#pragma unroll
    for (int k = 0; k < TAGS; ++k) g[k] = Gbuf[nb][k][p];
  }
  __syncthreads();
  if (w < 2) {                               // waves 0,1 cover all 64 columns
    float* go = G + (size_t)c * TAGS * TAGS;
#pragma unroll
    for (int k = 0; k < TAGS; ++k) go[k * TAGS + p] = g[k];
  }
}

// ---------------------------------------------------------------------------
// K2: sequential scan of fv through the chunk matrices; records fv at each
// chunk start, terminal score + argmax -> d_out[0], best_last.
// ---------------------------------------------------------------------------
__global__ __launch_bounds__(64) void k2_scan(
    const float* __restrict__ trans, const float* __restrict__ G,
    float* __restrict__ fvs, float* __restrict__ out_score,
    int* __restrict__ best_last) {
  const int n = threadIdx.x;  // 64 threads = 2 waves
  __shared__ float fv[TAGS];
  __shared__ float term[TAGS];
  fv[n] = (n == START_IDX) ? 0.0f : NEGVAL;
  __syncthreads();
  for (int c = 0; c < NCHUNK; ++c) {
    fvs[c * TAGS + n] = fv[n];
    const float* __restrict__ Gr = G + (size_t)c * TAGS * TAGS + n * TAGS;
    float m = NINF;
#pragma unroll
    for (int q = 0; q < TAGS; ++q) m = fmaxf(m, Gr[q] + fv[q]);
    __syncthreads();
    fv[n] = m;
    __syncthreads();
  }
  term[n] = fv[n] + trans[STOP_IDX * TAGS + n];
  __syncthreads();
  if (n == 0) {
    float best = term[0];
    int bi = 0;
#pragma unroll
    for (int i = 1; i < TAGS; ++i) {
      if (term[i] > best) { best = term[i]; bi = i; }
    }
    out_score[0] = best;
    best_last[0] = bi;
  }
}

// ---------------------------------------------------------------------------
// K3: exact backpointer recompute per chunk. One wave per chunk.
// scores^T tile D_{ij}[m,n'] = fv[16i+m] + T[16j+n', 16i+m] computed by
// v_wmma_f32_16x16x4_f32 with A = fv segment (k=0 column), B = ones,
// C = T^T tile preloaded in C/D layout. prev dim = VGPR index -> per-lane
// compare/select argmax, single shfl_xor(16) combines the two M-halves.
// Feats staged into LDS by the Tensor Data Mover (TENSORcnt); falls back to
// gfx1250 async global->LDS loads (ASYNCcnt) if the builtin is absent.
// ---------------------------------------------------------------------------
__global__ __launch_bounds__(32) void k3_backptr(
    const float* __restrict__ feats, const float* __restrict__ trans,
    const float* __restrict__ fvs, u8* __restrict__ bp) {
  const int c    = blockIdx.x;
  const int lane = threadIdx.x;      // 0..31, wave32
  const int nn   = lane & 15;
  const bool hi  = lane >= 16;

  __shared__ float sfeat[CHUNK * TAGS];  // 32 KB

  const unsigned ldsbase = (unsigned)(uintptr_t)(&sfeat[0]);
  const uint64_t gsrc =
      (uint64_t)(uintptr_t)(feats + (size_t)c * CHUNK * TAGS);

#if __has_builtin(__builtin_amdgcn_tensor_load_to_lds)
  {
    // Tensor DMA descriptor (ISA 8.3/8.4): 1 valid descriptor, type=2,
    // data_size=4B, 1 x 8192-element tile (contiguous 32 KB copy).
    const unsigned ne = CHUNK * TAGS;  // 8192 elements
    u32x4 g0;
    g0.x = 1u;                                   // count=1, no gather
    g0.y = ldsbase;                              // lds_addr
    g0.z = (unsigned)gsrc;                       // global_addr[31:0]
    g0.w = (unsigned)((gsrc >> 32) & 0x1FFFFFFu) // global_addr[56:32]
           | (2u << 30);                         // type=2 ("image")
    i32x8 g1;
    g1[0] = (int)(2u << 16);       // workgroup_mask=0, data_size=4B
    g1[1] = (int)(ne << 16);       // tensor_dim0[15:0] in [63:48]
    g1[2] = (int)(1u << 16);       // tensor_dim0 hi=0, tensor_dim1=1
    g1[3] = (int)(ne << 16);       // tensor_dim1 hi=0, tile_dim0=8192
    g1[4] = 1;                     // tile_dim1=1, tile_dim2=0
    g1[5] = (int)ne;               // tensor_dim0_stride lo
    g1[6] = 0;                     // stride hi / dim1_stride lo
    g1[7] = 0;
    i32x4 z4 = {0, 0, 0, 0};
#if defined(__clang_major__) && (__clang_major__ >= 23)
    i32x8 z8 = {0, 0, 0, 0, 0, 0, 0, 0};
    __builtin_amdgcn_tensor_load_to_lds(g0, g1, z4, z4, z8, 0);
#else
    __builtin_amdgcn_tensor_load_to_lds(g0, g1, z4, z4, 0);
#endif
  }
#else
  {
    const int nvec = (CHUNK * TAGS * 4) / 16;  // 2048 x 16B
    for (int r = lane; r < nvec; r += 32) {
      unsigned d = ldsbase + (unsigned)r * 16u;
      uint64_t a = gsrc + (uint64_t)r * 16u;
      asm volatile("global_load_async_to_lds_b128 %0, %1, off"
                   :: "v"(d), "v"(a) : "memory");
    }
  }
#endif

  // Preload transitions as scores^T C-tiles:
  // Ct[j][i][v]: row M = v(+8 for hi lanes) = prev%16, col N = nn = next%16.
  v8f Ct[4][4];
#pragma unroll
  for (int j = 0; j < 4; ++j)
#pragma unroll
    for (int i = 0; i < 4; ++i) {
      const float* tp = trans + (16 * j + nn) * TAGS + 16 * i + (hi ? 8 : 0);
#pragma unroll
      for (int v = 0; v < 8; ++v) Ct[j][i][v] = tp[v];
    }

  // A operands: fv segment in K=0 column (lanes 0..15), zero elsewhere.
  v2f A[4];
  const float* fv0 = fvs + c * TAGS;
#pragma unroll
  for (int i = 0; i < 4; ++i) {
    float f = fv0[16 * i + nn];
    A[i].x = hi ? 0.0f : f;
    A[i].y = 0.0f;
  }
  v2f ONES;
  ONES.x = 1.0f;
  ONES.y = 1.0f;

#if __has_builtin(__builtin_amdgcn_tensor_load_to_lds)
#if __has_builtin(__builtin_amdgcn_s_wait_tensorcnt)
  __builtin_amdgcn_s_wait_tensorcnt(0);
#else
  asm volatile("s_wait_tensorcnt 0" ::: "memory");
#endif
#else
  asm volatile("s_wait_asynccnt 0" ::: "memory");
#endif

  u8* bpc = bp + (size_t)c * CHUNK * TAGS;
  for (int s = 0; s < CHUNK; ++s) {
    float nf[4];
#pragma unroll
    for (int j = 0; j < 4; ++j) {
      float bestV = FLOWEST;
      int bestI = 0;
#pragma unroll
      for (int i = 0; i < 4; ++i) {
        v8f D = __builtin_amdgcn_wmma_f32_16x16x4_f32(
            false, A[i], false, ONES, (short)0, Ct[j][i], false, false);
#pragma unroll
        for (int v = 0; v < 8; ++v) {
          float sc = D[v];
          int idx = 16 * i + v + (hi ? 8 : 0);  // ascending per lane
          bool gt = sc > bestV;                 // strict > keeps first max
          bestV = gt ? sc : bestV;              // cmp + 2x cndmask, no canon
          bestI = gt ? idx : bestI;
        }
      }
      // Combine the two M-halves (lane l <-> l+16); ties -> smaller prev idx.
      float oV = __shfl_xor(bestV, 16, 32);
      int   oI = __shfl_xor(bestI, 16, 32);
      bool take = (oV > bestV) || ((oV == bestV) && (oI < bestI));
      bestV = take ? oV : bestV;
      bestI = take ? oI : bestI;

      float ft = sfeat[s * TAGS + 16 * j + nn];
      nf[j] = bestV + ft;
      if (!hi) bpc[s * TAGS + 16 * j + nn] = (u8)bestI;
    }
#pragma unroll
    for (int j = 0; j < 4; ++j) {  // fv_new already in A layout (seg j)
      A[j].x = hi ? 0.0f : nf[j];
      A[j].y = 0.0f;
    }
  }
}

// ---------------------------------------------------------------------------
// K4: per-chunk backtrace hypotheses: for each possible end tag y, the full
// in-chunk path and the chunk-entry tag (map). The chunk's 8 KB bp slab is
// staged into LDS with async loads so the dependent chase hits LDS latency.
// ---------------------------------------------------------------------------
__global__ __launch_bounds__(64) void k4_hyp(
    const u8* __restrict__ bp, u8* __restrict__ hyp, u8* __restrict__ map) {
  const int c = blockIdx.x;
  const int y = threadIdx.x;  // 64 hypotheses = 2 waves
  const size_t base = (size_t)c * CHUNK * TAGS;

  __shared__ u8 sbp[CHUNK * TAGS];  // 8 KB

  {
    unsigned ldsb = (unsigned)(uintptr_t)(&sbp[0]);
    uint64_t src = (uint64_t)(uintptr_t)(bp + base);
    const int nvec = (CHUNK * TAGS) / 16;  // 512 x 16B
    for (int r = y; r < nvec; r += 64) {
      unsigned d = ldsb + (unsigned)r * 16u;
      uint64_t a = src + (uint64_t)r * 16u;
      asm volatile("global_load_async_to_lds_b128 %0, %1, off"
                   :: "v"(d), "v"(a) : "memory");
    }
    asm volatile("s_wait_asynccnt 0" ::: "memory");
    __syncthreads();  // both waves' async copies complete
  }

  int carry = y;
  for (int tl = CHUNK - 1; tl >= 0; --tl) {
    hyp[base + (size_t)tl * TAGS + y] = (u8)carry;
    carry = sbp[tl * TAGS + carry];
  }
  map[c * TAGS + y] = (u8)carry;
}

// ---------------------------------------------------------------------------
// K5: stitch chunk maps right-to-left (1024 dependent lookups instead of
// 131072): ecarr[c] = tag entering chunk c from the right.
// ---------------------------------------------------------------------------
__global__ void k5_stitch(const int* __restrict__ best_last,
                          const u8* __restrict__ map, u8* __restrict__ ecarr) {
  if (threadIdx.x == 0 && blockIdx.x == 0) {
    int e = best_last[0];
    for (int c = NCHUNK - 1; c >= 0; --c) {
      ecarr[c] = (u8)e;
      e = map[c * TAGS + e];
    }
  }
}

// ---------------------------------------------------------------------------
// K6: parallel path emission: path[t] = hyp[t][ecarr[t/CHUNK]], as float.
// ---------------------------------------------------------------------------
__global__ __launch_bounds__(256) void k6_emit(
    const u8* __restrict__ hyp, const u8* __restrict__ ecarr,
    float* __restrict__ out_path) {
  int t = blockIdx.x * 256 + threadIdx.x;
  if (t < SEQ_LEN) {
    int e = ecarr[t / CHUNK];
    out_path[t] = (float)hyp[(size_t)t * TAGS + e];
  }
}

// ---------------------------------------------------------------------------
extern "C" void kernel_launch(void* const* d_in, const int* in_sizes, int n_in,
                              void* d_out, int out_size, void* d_ws,
                              size_t ws_size, hipStream_t stream) {
  (void)in_sizes; (void)n_in; (void)out_size;
  const float* feats = (const float*)d_in[0];  // (L, 64) f32
  const float* trans = (const float*)d_in[1];  // (64, 64) f32
  float* out = (float*)d_out;                  // [0]=score, [1..L]=path

  // Workspace layout (16B aligned regions).
  const size_t OFF_G    = 0;                                     // 16 MB
  const size_t OFF_FVS  = OFF_G    + (size_t)NCHUNK * TAGS * TAGS * 4;
  const size_t OFF_BP   = OFF_FVS  + (size_t)NCHUNK * TAGS * 4;  // 8 MB
  const size_t OFF_HYP  = OFF_BP   + (size_t)SEQ_LEN * TAGS;     // 8 MB
  const size_t OFF_MAP  = OFF_HYP  + (size_t)SEQ_LEN * TAGS;
  const size_t OFF_EC   = OFF_MAP  + (size_t)NCHUNK * TAGS;
  const size_t OFF_BL   = OFF_EC   + (size_t)NCHUNK;
  const size_t NEEDED   = OFF_BL + 16;
  if (ws_size < NEEDED) return;  // leave d_out poisoned rather than corrupt

  char* ws = (char*)d_ws;
  float* G    = (float*)(ws + OFF_G);
  float* fvs  = (float*)(ws + OFF_FVS);
  u8*    bp   = (u8*)(ws + OFF_BP);
  u8*    hyp  = (u8*)(ws + OFF_HYP);
  u8*    map  = (u8*)(ws + OFF_MAP);
  u8*    ec   = (u8*)(ws + OFF_EC);
  int*   bl   = (int*)(ws + OFF_BL);

  k1_chunk_gemm<<<NCHUNK, 256, 0, stream>>>(feats, trans, G);
  k2_scan<<<1, 64, 0, stream>>>(trans, G, fvs, out, bl);
  k3_backptr<<<NCHUNK, 32, 0, stream>>>(feats, trans, fvs, bp);
  k4_hyp<<<NCHUNK, 64, 0, stream>>>(bp, hyp, map);
  k5_stitch<<<1, 32, 0, stream>>>(bl, map, ec);
  k6_emit<<<SEQ_LEN / 256, 256, 0, stream>>>(hyp, ec, out + 1);
}